// SignedGCN_37675453120772
// MI455X (gfx1250) — compile-verified
//
#include <hip/hip_runtime.h>
#include <hip/hip_bf16.h>

typedef __attribute__((ext_vector_type(2))) float v2f;
typedef __attribute__((ext_vector_type(8))) float v8f;

#define DIM 128   // feature dim into both layers

// ---------------------------------------------------------------------------
// Kernel 1: per-edge in-degree count + weighted-degree (denominator) sum.
// Only cheap int/float atomics on N=50k counters (avg 16 collisions each).
// ---------------------------------------------------------------------------
__global__ __launch_bounds__(256) void gin_count_denom(
    const int* __restrict__ dst, const float* __restrict__ w,
    int* __restrict__ count, float* __restrict__ denom, int E) {
  int e = blockIdx.x * blockDim.x + threadIdx.x;
  if (e < E) {
    int d = dst[e];
    atomicAdd(&count[d], 1);
    atomicAdd(&denom[d], w[e]);
  }
}

// ---------------------------------------------------------------------------
// Kernel 2: single-block exclusive scan of counts -> CSR offsets, plus the
// per-node fused scale 1/(max(deg,1)*max(denom,1e-12)).
// ---------------------------------------------------------------------------
__global__ __launch_bounds__(1024) void gin_scan(
    const int* __restrict__ count, const float* __restrict__ denom,
    int* __restrict__ offset, int* __restrict__ cursor,
    float* __restrict__ scale, int n) {
  __shared__ int sdata[1024];
  __shared__ int s_carry;
  const int tid = threadIdx.x;
  if (tid == 0) s_carry = 0;
  __syncthreads();
  for (int base = 0; base < n; base += 1024) {
    int i = base + tid;
    int v = (i < n) ? count[i] : 0;
    sdata[tid] = v;
    __syncthreads();
    #pragma unroll
    for (int off = 1; off < 1024; off <<= 1) {
      int t = (tid >= off) ? sdata[tid - off] : 0;
      __syncthreads();
      sdata[tid] += t;
      __syncthreads();
    }
    int incl = sdata[tid];
    int chunkTotal = sdata[1023];
    int excl = incl - v + s_carry;
    if (i < n) {
      offset[i] = excl;
      cursor[i] = excl;
      float dg = (float)v;
      float sc = 1.0f / ((dg < 1.0f ? 1.0f : dg) * fmaxf(denom[i], 1e-12f));
      scale[i] = sc;
      if (i == n - 1) offset[n] = excl + v;   // total = E
    }
    __syncthreads();
    if (tid == 0) s_carry += chunkTotal;
    __syncthreads();
  }
}

// ---------------------------------------------------------------------------
// Kernel 3: bucket fill — scatter edge ids into CSR order (int atomics only).
// ---------------------------------------------------------------------------
__global__ __launch_bounds__(256) void gin_fill(
    const int* __restrict__ dst, int* __restrict__ cursor,
    int* __restrict__ eidx, int E) {
  int e = blockIdx.x * blockDim.x + threadIdx.x;
  if (e < E) {
    int pos = atomicAdd(&cursor[dst[e]], 1);
    eidx[pos] = e;
  }
}

// ---------------------------------------------------------------------------
// Kernel 4: gather-aggregate. One wave32 per destination node; each lane owns
// a float4 slice of the 128-wide feature row. Edge metadata loaded coalesced
// (one edge per lane per 32-edge chunk) and broadcast via __shfl. Produces
// t[n,:] = x[n,:] + scale[n] * sum_e w_e * x[src_e,:]  -- no float atomics.
// ---------------------------------------------------------------------------
__global__ __launch_bounds__(256) void gin_aggregate(
    const float* __restrict__ xin, const float* __restrict__ w,
    const int* __restrict__ src, const int* __restrict__ eidx,
    const int* __restrict__ offset, const float* __restrict__ scale,
    float* __restrict__ tout, int n_nodes) {
  int wave = (int)((blockIdx.x * blockDim.x + threadIdx.x) >> 5);
  int lane = threadIdx.x & 31;
  if (wave >= n_nodes) return;
  int beg = offset[wave];
  int end = offset[wave + 1];
  float4 acc = make_float4(0.f, 0.f, 0.f, 0.f);
  for (int base = beg; base < end; base += 32) {
    int slot = base + lane;
    int e  = (slot < end) ? eidx[slot] : 0;
    int s  = (slot < end) ? src[e]     : 0;
    float ww = (slot < end) ? w[e]     : 0.f;
    int cnt = end - base; if (cnt > 32) cnt = 32;
    for (int j = 0; j < cnt; ++j) {
      int   sj = __shfl(s,  j, 32);
      float wj = __shfl(ww, j, 32);
      const float4* row = (const float4*)(xin + (size_t)sj * DIM);
      float4 v = row[lane];
      acc.x += v.x * wj; acc.y += v.y * wj;
      acc.z += v.z * wj; acc.w += v.w * wj;
    }
  }
  float sc = scale[wave];
  const float4* xrow = (const float4*)(xin + (size_t)wave * DIM);
  float4 xv = xrow[lane];
  float4 o;
  o.x = xv.x + acc.x * sc; o.y = xv.y + acc.y * sc;
  o.z = xv.z + acc.z * sc; o.w = xv.w + acc.w * sc;
  ((float4*)(tout + (size_t)wave * DIM))[lane] = o;
}

// ---------------------------------------------------------------------------
// Kernel 5: WMMA GEMM + bias + ReLU.   out = relu(t @ W.T + b)
// Full fp32 via V_WMMA_F32_16X16X4_F32 (D=A*B+C, 16x16 tile, K-step 4).
// A-tile staged in LDS (padded stride 132 -> conflict-free float2 operand
// reads); B (weights) read from global (L1/L2 resident, <=64KB).
//   A operand (32b, 16x4):  lane m=lane&15 holds K = kb + 2*(lane>>4) + {0,1}
//   B operand mirrored; C/D: VGPR r = row r (+8 for lanes 16-31), col=lane&15
// ---------------------------------------------------------------------------
template <int DOUT>
__global__ __launch_bounds__(256) void gin_gemm_bias_relu(
    const float* __restrict__ tin, const float* __restrict__ Wm,
    const float* __restrict__ bias, float* __restrict__ out, int n) {
  constexpr int K    = DIM;
  constexpr int COLT = DOUT / 16;   // 16-col tiles
  constexpr int NW   = 8;           // waves per block
  constexpr int ROWT = NW / COLT;   // 16-row tiles per block
  constexpr int ROWS = ROWT * 16;
  constexpr int PAD  = 132;         // padded LDS row stride (floats)
  __shared__ float sA[ROWS * PAD];

  const int tid = threadIdx.x;
  const int rowBase = blockIdx.x * ROWS;

  // cooperative float4 load of the A tile (OOB rows -> 0)
  for (int f = tid; f < ROWS * (K / 4); f += 256) {
    int r  = f / (K / 4);
    int c4 = f % (K / 4);
    int grow = rowBase + r;
    float4 v = make_float4(0.f, 0.f, 0.f, 0.f);
    if (grow < n) v = *(const float4*)&tin[(size_t)grow * K + c4 * 4];
    *(float4*)&sA[r * PAD + c4 * 4] = v;   // 528B row stride: 16B aligned
  }
  __syncthreads();

  const int wv = tid >> 5, lane = tid & 31;
  const int colTile = wv % COLT, rowTile = wv / COLT;
  const int m    = lane & 15;
  const int koff = (lane >> 4) << 1;
  const float* sArow = &sA[(rowTile * 16 + m) * PAD];
  const float* Brow  = &Wm[(size_t)(colTile * 16 + m) * K];

  v8f c = {};
  #pragma unroll
  for (int k = 0; k < K; k += 4) {
    v2f a = *(const v2f*)&sArow[k + koff];
    v2f b = *(const v2f*)&Brow[k + koff];
    c = __builtin_amdgcn_wmma_f32_16x16x4_f32(
        /*neg_a=*/false, a, /*neg_b=*/false, b,
        /*c_mod=*/(short)0, c, /*reuse_a=*/false, /*reuse_b=*/false);
  }

  const int col = colTile * 16 + m;
  const float bb = bias[col];
  const int gRow0 = rowBase + rowTile * 16 + ((lane >> 4) * 8);
  #pragma unroll
  for (int r = 0; r < 8; ++r) {
    int grow = gRow0 + r;
    if (grow < n) {
      float v = c[r] + bb;
      out[(size_t)grow * DOUT + col] = v > 0.f ? v : 0.f;
    }
  }
}

// ---------------------------------------------------------------------------
extern "C" void kernel_launch(void* const* d_in, const int* in_sizes, int n_in,
                              void* d_out, int out_size, void* d_ws, size_t ws_size,
                              hipStream_t stream) {
  const float* x  = (const float*)d_in[0];   // [N,128]
  const float* w  = (const float*)d_in[1];   // [E]
  const float* W1 = (const float*)d_in[2];   // [128,128]
  const float* b1 = (const float*)d_in[3];   // [128]
  const float* W2 = (const float*)d_in[4];   // [64,128]
  const float* b2 = (const float*)d_in[5];   // [64]
  const int* src  = (const int*)d_in[6];     // [E]
  const int* dst  = (const int*)d_in[7];     // [E]
  const int N = in_sizes[0] / DIM;
  const int E = in_sizes[1];

  // carve 256B-aligned workspace
  char* ws = (char*)d_ws;
  size_t off = 0;
  auto carve = [&](size_t bytes) -> void* {
    void* p = ws + off;
    off += (bytes + 255) & ~(size_t)255;
    return p;
  };
  int*   count  = (int*)  carve((size_t)N * 4);
  float* denom  = (float*)carve((size_t)N * 4);
  int*   offset = (int*)  carve((size_t)(N + 1) * 4);
  int*   cursor = (int*)  carve((size_t)N * 4);
  float* scale  = (float*)carve((size_t)N * 4);
  int*   eidx   = (int*)  carve((size_t)E * 4);
  float* tbuf   = (float*)carve((size_t)N * DIM * 4);  // GEMM input
  float* hbuf   = (float*)carve((size_t)N * DIM * 4);  // layer-1 output
  (void)ws_size;

  const int edgeBlocks = (E + 255) / 256;
  const int aggBlocks  = (N + 7) / 8;          // 8 waves/block, 1 node/wave

  // ---- graph structure (shared by both layers) ----
  hipMemsetAsync(count, 0, (size_t)N * 4, stream);
  hipMemsetAsync(denom, 0, (size_t)N * 4, stream);
  gin_count_denom<<<edgeBlocks, 256, 0, stream>>>(dst, w, count, denom, E);
  gin_scan<<<1, 1024, 0, stream>>>(count, denom, offset, cursor, scale, N);
  gin_fill<<<edgeBlocks, 256, 0, stream>>>(dst, cursor, eidx, E);

  // ---- layer 1: aggregate(x) -> t; h = relu(t @ W1.T + b1) ----
  gin_aggregate<<<aggBlocks, 256, 0, stream>>>(x, w, src, eidx, offset, scale, tbuf, N);
  gin_gemm_bias_relu<128><<<(N + 15) / 16, 256, 0, stream>>>(tbuf, W1, b1, hbuf, N);

  // ---- layer 2: aggregate(h) -> t; out = relu(t @ W2.T + b2) ----
  gin_aggregate<<<aggBlocks, 256, 0, stream>>>(hbuf, w, src, eidx, offset, scale, tbuf, N);
  gin_gemm_bias_relu<64><<<(N + 31) / 32, 256, 0, stream>>>(tbuf, W2, b2, (float*)d_out, N);
}